// FastText_33956011442351
// MI455X (gfx1250) — compile-verified
//
#include <hip/hip_runtime.h>
#include <hip/hip_bf16.h>

typedef __attribute__((ext_vector_type(2))) float v2f;
typedef __attribute__((ext_vector_type(8))) float v8f;

namespace {
constexpr int kB   = 65536;   // batch rows
constexpr int kPos = 6;
constexpr int kNeg = 30;
constexpr int kTot = 36;      // pos + neg scores per row
constexpr int kNgr = 12;
constexpr int kD   = 50;      // embedding dim (padded to 52 = 13 K-chunks of 4)
}

__device__ __forceinline__ float softplus_f(float x) {
  // log(1 + exp(x)), numerically stable
  float ax = fabsf(x);
  return fmaxf(x, 0.f) + __logf(1.f + __expf(-ax));
}

// One wave (32 lanes) handles one batch row. 8 waves per 256-thread block.
__global__ __launch_bounds__(256) void fasttext_rows(
    const int* __restrict__ in_lbl,
    const int* __restrict__ pos_lbl,
    const int* __restrict__ neg_lbl,
    const int* __restrict__ tri_idx,
    const int* __restrict__ ngr_mask,
    const float* __restrict__ centerW,
    const float* __restrict__ bgW,
    const float* __restrict__ triW,
    float* __restrict__ partial)
{
  const int lane = threadIdx.x & 31;
  const int wid  = threadIdx.x >> 5;
  const int b    = blockIdx.x * 8 + wid;
  const int m16  = lane & 15;   // row index within a 16-wide k-tile
  const int h    = lane >> 4;   // K half: lanes 0-15 -> K=0,1 ; lanes 16-31 -> K=2,3

  // ---- m[d] = center[d] + sum_n mask_n * trigram_n[d]; lane L owns d=2L,2L+1 ----
  float mx = 0.f, my = 0.f;
  {
    const int d0   = 2 * lane;
    const bool own = (d0 + 1) < kD;            // lanes 0..24 real; lane 25 = pad zeros
    const int cl = in_lbl[b];
    if (own) {
      const float* p = centerW + (long)cl * kD + d0;
      mx = p[0]; my = p[1];
    }
    #pragma unroll
    for (int n = 0; n < kNgr; ++n) {
      const int msk = ngr_mask[b * kNgr + n];  // wave-uniform -> skips load when 0
      if (msk != 0) {
        const int ti = tri_idx[b * kNgr + n];
        if (own) {
          const float* p = triW + (long)ti * kD + d0;
          mx += p[0]; my += p[1];
        }
      }
    }
  }

  // ---- per-lane A-matrix row pointers: tile t covers scores k = 16t + m16.
  //      Padded k (>= 36) aliases row 0: always in-bounds, finite, L0-hot,
  //      masked out at the softplus stage -> no divergent branches in the loop.
  const float* rowp[3];
  #pragma unroll
  for (int t = 0; t < 3; ++t) {
    const int k = 16 * t + m16;
    int lbl = 0;
    if (k < kPos)      lbl = pos_lbl[b * kPos + k];
    else if (k < kTot) lbl = neg_lbl[b * kNeg + (k - kPos)];
    rowp[t] = bgW + (long)lbl * kD;
  }

  // ---- scores via V_WMMA_F32_16X16X4_F32, K accumulated over 13 chunks ----
  // A[M,K]: M = m16 (embedding row), K-pair selected by lane half h.
  // B[K,N]: m-chunk value, identical across all 16 N columns.
  v8f acc[3] = {};
  #pragma unroll
  for (int c = 0; c < 13; ++c) {
    const int src = 2 * c + h;                 // lane that owns m[4c+2h], m[4c+2h+1]
    v2f bv;
    bv.x = __shfl(mx, src, 32);
    bv.y = __shfl(my, src, 32);
    int doff = 4 * c + 2 * h;
    if (doff > kD - 2) doff = kD - 2;          // clamp pad dims (paired with m==0)
    #pragma unroll
    for (int t = 0; t < 3; ++t) {
      const float* p = rowp[t] + doff;
      v2f av; av.x = p[0]; av.y = p[1];        // unconditional, always in-bounds
      acc[t] = __builtin_amdgcn_wmma_f32_16x16x4_f32(
          false, av, false, bv, (short)0, acc[t], false, false);
    }
  }

  // ---- softplus + masked sum. All 16 N-columns of D are duplicates, so only
  //      lanes with N==0 contribute: lane 0 (rows 0..7), lane 16 (rows 8..15).
  float local = 0.f;
  if (m16 == 0) {
    #pragma unroll
    for (int t = 0; t < 3; ++t) {
      #pragma unroll
      for (int v = 0; v < 8; ++v) {
        const int k = 16 * t + 8 * h + v;      // D layout: M = 8*(lane/16) + vgpr
        if (k < kTot) {
          const float s = acc[t][v];
          local += softplus_f((k < kPos) ? -s : s);
        }
      }
    }
  }

  // Only lanes 0 and 16 are nonzero: one xor-16 exchange combines them.
  local += __shfl_xor(local, 16, 32);
  if (lane == 0) partial[b] = local;           // deterministic per-row partial
}

// Fixed-shape deterministic tree reduction: one block sums n_per_block inputs.
__global__ __launch_bounds__(256) void reduce256(
    const float* __restrict__ in, float* __restrict__ out, int n_per_block)
{
  __shared__ float sm[8];
  const int tid  = threadIdx.x;
  const int base = blockIdx.x * n_per_block;
  float s = 0.f;
  for (int i = tid; i < n_per_block; i += 256) s += in[base + i];
  #pragma unroll
  for (int off = 16; off > 0; off >>= 1) s += __shfl_xor(s, off, 32);
  if ((tid & 31) == 0) sm[tid >> 5] = s;
  __syncthreads();
  if (tid == 0) {
    float t = 0.f;
    #pragma unroll
    for (int w = 0; w < 8; ++w) t += sm[w];
    out[blockIdx.x] = t;
  }
}

extern "C" void kernel_launch(void* const* d_in, const int* in_sizes, int n_in,
                              void* d_out, int out_size, void* d_ws, size_t ws_size,
                              hipStream_t stream) {
  const int*   in_lbl  = (const int*)d_in[0];
  const int*   pos_lbl = (const int*)d_in[1];
  const int*   neg_lbl = (const int*)d_in[2];
  const int*   tri_idx = (const int*)d_in[3];
  const int*   ngr_msk = (const int*)d_in[4];
  const float* centerW = (const float*)d_in[5];
  const float* bgW     = (const float*)d_in[6];
  const float* triW    = (const float*)d_in[7];
  float* ws  = (float*)d_ws;   // [0,65536): row partials; [65536,65792): stage-2
  float* out = (float*)d_out;

  fasttext_rows<<<kB / 8, 256, 0, stream>>>(in_lbl, pos_lbl, neg_lbl, tri_idx,
                                            ngr_msk, centerW, bgW, triW, ws);
  reduce256<<<256, 256, 0, stream>>>(ws, ws + kB, kB / 256);
  reduce256<<<1, 256, 0, stream>>>(ws + kB, out, 256);
}